// TiledFeaturesMap_35107062677846
// MI455X (gfx1250) — compile-verified
//
#include <hip/hip_runtime.h>
#include <stdint.h>

// ---------------- problem constants (match reference) ----------------
#define B_     4
#define N_     2048
#define C_     512
#define TSZ    9
#define TSTEP  6
#define TCUT   2
#define MT     21          // max tiles per dim: (120)/6 + 1
#define MTT    (MT*MT)     // 441
#define MAPD   130         // map buffer dim (H,W <= 129)
#define CHUNK  128         // channels per gather block
#define NCHUNK (C_/CHUNK)  // 4
#define RS     132         // LDS row stride in floats (128 + 4 pad -> 16B aligned, 2-way max conflict)

// ---------------- ws layout (int32 units) ----------------
#define OFF_BBOX 0                      // [B][4] : ymin,ymax,xmin,xmax
#define OFF_ANY  (OFF_BBOX + B_*4)      // [B]    : any center-occupied tile
#define OFF_K    (OFF_ANY + B_)         // [1]
#define OFF_MAP  (OFF_K + 7)            // [B][130][130] point index or -1
#define MAPN     (B_*MAPD*MAPD)
#define OFF_CF   (OFF_MAP + MAPN)       // [B*MTT] center-occupancy flag
#define OFF_FF   (OFF_CF + B_*MTT)      // [B*MTT] full-occupancy flag
#define OFF_SLOT (OFF_FF + B_*MTT)      // [B*MTT] output slot or -1
#define WS_INTS  (OFF_SLOT + B_*MTT)

// ---------------- CDNA5 async global<->LDS helpers ----------------
typedef int v4i_ __attribute__((ext_vector_type(4)));
typedef __attribute__((address_space(1))) v4i_* gv4_t;   // global int4*
typedef __attribute__((address_space(3))) v4i_* lv4_t;   // LDS int4*
typedef __attribute__((address_space(1))) int*  gi_t;    // global int*
typedef __attribute__((address_space(3))) int*  li_t;    // LDS int*

__device__ __forceinline__ void async_load_b128(const float* g, float* l) {
#if defined(__gfx1250__) && __has_builtin(__builtin_amdgcn_global_load_async_to_lds_b128)
  __builtin_amdgcn_global_load_async_to_lds_b128(
      (gv4_t)const_cast<float*>(g), (lv4_t)l, /*offset=*/0, /*cpol=*/0);
#else
  *(float4*)l = *(const float4*)g;      // fallback: synchronous 16B copy via VGPRs
#endif
}

// Direct LDS -> global streaming store, 4B per lane, arbitrary per-lane LDS addr.
// HAVE_ASYNC_STORE is 1 if the async path is used (else the DS-read fallback is compiled).
#if defined(__gfx1250__) && __has_builtin(__builtin_amdgcn_global_store_async_from_lds_b32)
#define HAVE_ASYNC_STORE 1
#else
#define HAVE_ASYNC_STORE 0
#endif

__device__ __forceinline__ void async_store_b32(float* g, const float* l) {
#if HAVE_ASYNC_STORE
  __builtin_amdgcn_global_store_async_from_lds_b32(
      (gi_t)g, (li_t)const_cast<float*>(l), /*offset=*/0, /*cpol=*/0);
#else
  *g = *l;                               // fallback: LDS read + global store via VGPR
#endif
}

__device__ __forceinline__ void wait_async() {
#if defined(__gfx1250__) && __has_builtin(__builtin_amdgcn_s_wait_asynccnt)
  __builtin_amdgcn_s_wait_asynccnt(0);
#elif defined(__gfx1250__)
  asm volatile("s_wait_asynccnt 0" ::: "memory");
#endif
}

// ---------------- kernel 0: init workspace ----------------
__global__ void k_init(int* __restrict__ ws) {
  int i = blockIdx.x * blockDim.x + threadIdx.x;
  if (i >= WS_INTS) return;
  int v = 0;
  if (i < OFF_ANY) {                 // bbox: min fields -> INT_MAX, max fields -> INT_MIN
    int f = i & 3;
    v = (f == 0 || f == 2) ? 0x7fffffff : (int)0x80000000;
  } else if (i >= OFF_MAP && i < OFF_MAP + MAPN) {
    v = -1;
  }
  ws[i] = v;
}

// ---------------- kernel 1: per-batch bbox via atomics ----------------
__global__ void k_bbox(const int* __restrict__ ys, const int* __restrict__ xs,
                       int* __restrict__ ws) {
  int t = blockIdx.x * blockDim.x + threadIdx.x;
  if (t >= B_ * N_) return;
  int b = t / N_;
  int y = ys[t];
  if (y > -1) {
    int x = xs[t];
    atomicMin(&ws[OFF_BBOX + b*4 + 0], y);
    atomicMax(&ws[OFF_BBOX + b*4 + 1], y);
    atomicMin(&ws[OFF_BBOX + b*4 + 2], x);
    atomicMax(&ws[OFF_BBOX + b*4 + 3], x);
  }
}

// ---------------- kernel 2: scatter point index into dense map ----------------
__global__ void k_scatter(const int* __restrict__ ys, const int* __restrict__ xs,
                          int* __restrict__ ws) {
  int t = blockIdx.x * blockDim.x + threadIdx.x;
  if (t >= B_ * N_) return;
  int b = t / N_, n = t - b * N_;
  int y = ys[t];
  if (y > -1) {
    int x  = xs[t];
    int my = y - ws[OFF_BBOX + b*4 + 0];
    int mx = x - ws[OFF_BBOX + b*4 + 2];
    ws[OFF_MAP + (b*MAPD + my)*MAPD + mx] = n;  // cells unique per batch: no race
  }
}

// ---------------- kernel 3: per-tile occupancy flags ----------------
__global__ void k_flags(int* __restrict__ ws) {
  int t = blockIdx.x * blockDim.x + threadIdx.x;
  if (t >= B_ * MTT) return;
  int b = t / MTT, ij = t - b * MTT;
  int i = ij / MT, j = ij - i * MT;
  int ymin = ws[OFF_BBOX + b*4 + 0], ymax = ws[OFF_BBOX + b*4 + 1];
  int xmin = ws[OFF_BBOX + b*4 + 2], xmax = ws[OFF_BBOX + b*4 + 3];
  int nH = (ymax - ymin + 1) / TSTEP + 1;
  int nW = (xmax - xmin + 1) / TSTEP + 1;
  int cf = 0, ff = 0;
  if (i < nH && j < nW) {
    const int* mapb = ws + OFF_MAP + b*MAPD*MAPD;
    for (int h = 0; h < TSZ; ++h)
      for (int w = 0; w < TSZ; ++w) {
        if (mapb[(i*TSTEP + h)*MAPD + (j*TSTEP + w)] >= 0) {
          ff = 1;
          if (h >= TCUT && h < TSZ-TCUT && w >= TCUT && w < TSZ-TCUT) cf = 1;
        }
      }
    if (cf) atomicOr(&ws[OFF_ANY + b], 1);
  }
  ws[OFF_CF + t] = cf;
  ws[OFF_FF + t] = ff;
}

// ---------------- kernel 4: single-block scan -> slots + f_ns ----------------
__global__ __launch_bounds__(256)
void k_scan(int* __restrict__ ws, float* __restrict__ fns) {
  __shared__ int sh[256];
  const int tid = threadIdx.x;
  const int TOT = B_ * MTT;            // 1764
  const int ITEMS = 7;                 // 256*7 >= 1764
  int sel[ITEMS]; int cnt = 0;
  for (int k = 0; k < ITEMS; ++k) {
    int t = tid * ITEMS + k;
    int s = 0;
    if (t < TOT) {
      int b = t / MTT;
      s = ws[OFF_ANY + b] ? ws[OFF_CF + t] : ws[OFF_FF + t];  // reference fallback branch
    }
    sel[k] = s; cnt += s;
  }
  sh[tid] = cnt; __syncthreads();
  for (int off = 1; off < 256; off <<= 1) {   // Hillis-Steele inclusive scan
    int u = (tid >= off) ? sh[tid - off] : 0;
    __syncthreads();
    sh[tid] += u;
    __syncthreads();
  }
  int base = sh[tid] - cnt;                   // exclusive prefix
  for (int k = 0; k < ITEMS; ++k) {
    int t = tid * ITEMS + k;
    if (t < TOT) {
      if (sel[k]) {
        ws[OFF_SLOT + t] = base;
        fns[base] = (float)(t / MTT);         // f_ns = batch index
        ++base;
      } else {
        ws[OFF_SLOT + t] = -1;
      }
    }
  }
  if (tid == 255) ws[OFF_K] = sh[255];
}

// ---------------- kernel 5: tile gather, fully on the async LDS path ----------------
__global__ __launch_bounds__(256)
void k_gather(const float* __restrict__ feat, const int* __restrict__ ws,
              float* __restrict__ tiles) {
  __shared__ float lds[TSZ*TSZ * RS];         // 81 rows x 132 floats = 42768 B
  int blk   = blockIdx.x;
  int chunk = blk & (NCHUNK - 1);
  int t     = blk >> 2;                       // NCHUNK == 4
  int b     = t / MTT;
  int ij    = t - b * MTT;
  int i     = ij / MT, j = ij - i * MT;

  int ymin = ws[OFF_BBOX + b*4 + 0], ymax = ws[OFF_BBOX + b*4 + 1];
  int xmin = ws[OFF_BBOX + b*4 + 2], xmax = ws[OFF_BBOX + b*4 + 3];
  int nH = (ymax - ymin + 1) / TSTEP + 1;
  int nW = (xmax - xmin + 1) / TSTEP + 1;
  int slot = (i < nH && j < nW) ? ws[OFF_SLOT + t] : -1;
  if (slot < 0) return;                       // uniform across block

  const int tid  = threadIdx.x;
  const int wave = tid >> 5, lane = tid & 31;
  const int c0   = chunk * CHUNK;
  const int* mapb = ws + OFF_MAP + b*MAPD*MAPD;

  // Load phase: one wave-wide async b128 per occupied cell moves a whole
  // 512B channel-row chunk straight into LDS (32 lanes x 16B). Empty cells zero-filled.
  for (int cell = wave; cell < TSZ*TSZ; cell += 8) {
    int h = cell / TSZ, w = cell - h * TSZ;
    int n = mapb[(i*TSTEP + h)*MAPD + (j*TSTEP + w)];   // wave-uniform
    float* ldst = &lds[cell*RS + lane*4];               // 16B aligned (528B row stride)
    if (n >= 0) {
      const float* g = feat + ((size_t)b*N_ + (size_t)n)*C_ + c0 + lane*4;
      async_load_b128(g, ldst);
    } else {
      *(float4*)ldst = make_float4(0.f, 0.f, 0.f, 0.f);
    }
  }
  wait_async();          // each wave drains its own ASYNCcnt...
  __syncthreads();       // ...then all rows are visible block-wide

  // Store phase: direct LDS->global async stores do the transpose in one
  // instruction per 128B: global addresses contiguous across lanes, per-lane
  // LDS address = cell*RS + c. Tail split (10368 = 40.5*256) is whole-wave
  // uniform, so EXEC is all-ones for every issued async op.
  float* outB = tiles + (size_t)slot * (C_*TSZ*TSZ) + (size_t)c0 * (TSZ*TSZ);
  for (int idx = tid; idx < CHUNK * TSZ*TSZ; idx += 256) {
    int c    = idx / (TSZ*TSZ);
    int cell = idx - c * (TSZ*TSZ);
    async_store_b32(&outB[idx], &lds[cell*RS + c]);
  }
  // S_ENDPGM performs an implicit wait-idle; LDS is not reused after this point.
}

// ---------------- launcher ----------------
extern "C" void kernel_launch(void* const* d_in, const int* in_sizes, int n_in,
                              void* d_out, int out_size, void* d_ws, size_t ws_size,
                              hipStream_t stream) {
  const float* feat = (const float*)d_in[0];
  const int*   ys   = (const int*)d_in[1];
  const int*   xs   = (const int*)d_in[2];
  int*   ws   = (int*)d_ws;
  float* out  = (float*)d_out;

  // out_size = K*(1 + C*81)  ->  recover K on host (deterministic)
  int K = out_size / (C_ * TSZ * TSZ + 1);
  float* tiles = out + K;                     // f_ns occupies out[0..K)

  k_init   <<<(WS_INTS + 255)/256, 256, 0, stream>>>(ws);
  k_bbox   <<<(B_*N_)/256,         256, 0, stream>>>(ys, xs, ws);
  k_scatter<<<(B_*N_)/256,         256, 0, stream>>>(ys, xs, ws);
  k_flags  <<<(B_*MTT + 255)/256,  256, 0, stream>>>(ws);
  k_scan   <<<1,                   256, 0, stream>>>(ws, out);
  k_gather <<<B_*MTT*NCHUNK,       256, 0, stream>>>(feat, ws, tiles);
}